// Join_1580547966999
// MI455X (gfx1250) — compile-verified
//
#include <hip/hip_runtime.h>

// out[e] = [ unary[idx1[e]] (128f) | unary[idx2[e]] (128f) | binary[e] (64f) ]
//
// Pure data movement (0 FLOPs): ~2.57 GB per call -> ~110us floor at 23.3 TB/s HBM.
// Strategy (MI455X):
//   * one wave32 per edge: 128-float unary row == 32 x B128 (one lane each),
//     64-float binary row == 32 x B64 -- every access fully coalesced
//   * unary (51 MB) kept L2-resident (192 MB L2): gathers use default RT policy,
//     the 1.28 GB output stream is non-temporal so it never evicts the table
//   * binary passthrough rides the CDNA5 async LDS path
//     (global_load_async_to_lds_b64 / s_wait_asynccnt / global_store_async_from_lds_b64),
//     issued before the gather stores so the ASYNCcnt transfer overlaps them
//   * wave-uniform row indices -> v_readfirstlane -> scalar-base (saddr) gathers

typedef __attribute__((ext_vector_type(4))) float v4f;
typedef __attribute__((ext_vector_type(2))) float v2f;

__global__ __launch_bounds__(256) void join_edge_kernel(
    const v4f* __restrict__ unary,    // [u_nodes][32] float4
    const v2f* __restrict__ binary2,  // [edges][32]  float2
    const int* __restrict__ idx1,     // [edges]
    const int* __restrict__ idx2,     // [edges]
    v4f*       __restrict__ out4,     // [edges][80]  float4
    v2f*       __restrict__ out2,     // [edges][160] float2 (same buffer)
    long long edges)
{
    __shared__ v2f stage[256];        // 8B per lane staging for the async hop

    const int wave = threadIdx.x >> 5;      // 8 waves / block
    const int lane = threadIdx.x & 31;
    const long long edge = (long long)blockIdx.x * 8 + wave;
    if (edge >= edges) return;              // wave-uniform exit (EXEC all-0 or all-1)

    // Wave-uniform indices: 32 lanes hit one address (single coalesced request),
    // then readfirstlane pins the row in an SGPR -> saddr-form gathers.
    int row1 = __builtin_amdgcn_readfirstlane(idx1[edge]);
    int row2 = __builtin_amdgcn_readfirstlane(idx2[edge]);

    // Kick off the binary row copy through LDS first (ASYNCcnt path) so it
    // overlaps the gather latency below.
    unsigned lds_off = (unsigned)(unsigned long long)(&stage[threadIdx.x]);
    unsigned long long bsrc =
        (unsigned long long)(binary2 + edge * 32 + lane);
    unsigned long long bdst =
        (unsigned long long)(out2 + edge * 160 + 128 + lane);
    asm volatile("global_load_async_to_lds_b64 %0, %1, off"
                 :: "v"(lds_off), "v"(bsrc)
                 : "memory");

    // Two independent gather chains per lane; table stays in L2 (RT policy).
    v4f a = unary[(long long)row1 * 32 + lane];
    v4f b = unary[(long long)row2 * 32 + lane];

    // Streaming output: non-temporal so the unary table is never evicted.
    v4f* o = out4 + edge * 80 + lane;
    __builtin_nontemporal_store(a, o);
    __builtin_nontemporal_store(b, o + 32);

    // Binary row has landed in LDS by now (or we stall briefly); push it out.
    asm volatile("s_wait_asynccnt 0" ::: "memory");
    asm volatile("global_store_async_from_lds_b64 %0, %1, off"
                 :: "v"(bdst), "v"(lds_off)
                 : "memory");
    // s_endpgm performs an implicit wait-idle, covering the async store.
}

extern "C" void kernel_launch(void* const* d_in, const int* in_sizes, int n_in,
                              void* d_out, int out_size, void* d_ws, size_t ws_size,
                              hipStream_t stream)
{
    const float* unary  = (const float*)d_in[0];   // [u_nodes*128] f32
    const float* binary = (const float*)d_in[1];   // [edges*64]    f32
    const int*   idx1   = (const int*)d_in[2];     // [edges]
    const int*   idx2   = (const int*)d_in[3];     // [edges]
    float*       out    = (float*)d_out;           // [edges*320]   f32

    long long edges = (long long)in_sizes[2];
    int grid = (int)((edges + 7) / 8);             // 8 waves (edges) per block

    join_edge_kernel<<<grid, 256, 0, stream>>>(
        (const v4f*)unary, (const v2f*)binary, idx1, idx2,
        (v4f*)out, (v2f*)out, edges);
}